// CrfRnnLayerSP_3942779978351
// MI455X (gfx1250) — compile-verified
//
#include <hip/hip_runtime.h>
#include <math.h>

#define HH 256
#define WW 256
#define NPIX (HH*WW)
#define CC 6
#define RSP 9            // spatial Gaussian radius (3*theta_gamma)
#define RBW 4            // bilateral radius -> 9x9 = 81 taps
#define EPSF 1e-5f
#define TILE 16
#define BT (TILE + 2*RBW) // 24

typedef float v2f_t __attribute__((ext_vector_type(2)));
typedef float v8f_t __attribute__((ext_vector_type(8)));

#if defined(__has_builtin)
#if __has_builtin(__builtin_amdgcn_global_load_async_to_lds_b32) && \
    __has_builtin(__builtin_amdgcn_s_wait_asynccnt)
#define HAVE_ASYNC_LDS 1
#endif
#endif

// exp(-d^2 / (2*3^2)) for d = -9..9
__device__ const float GK[19] = {
    0.0111090f, 0.0285655f, 0.0657285f, 0.1353353f, 0.2493520f,
    0.4111120f, 0.6065307f, 0.8007374f, 0.9459595f, 1.0000000f,
    0.9459595f, 0.8007374f, 0.6065307f, 0.4111120f, 0.2493520f,
    0.1353353f, 0.0657285f, 0.0285655f, 0.0111090f};

// ------- prep: Msp = comp @ sk, Mbp = comp @ bk, padded to 16x8 (zeros) -----
__global__ void k_prep(const float* __restrict__ sk, const float* __restrict__ bk,
                       const float* __restrict__ comp,
                       float* __restrict__ Msp, float* __restrict__ Mbp) {
    int t = threadIdx.x;            // 0..127 -> 16 rows x 8 cols
    int i = t >> 3, j = t & 7;
    float a = 0.f, b = 0.f;
    if (i < CC && j < CC) {
        #pragma unroll
        for (int k = 0; k < 6; ++k) {
            a += comp[i*6+k] * sk[k*6+j];
            b += comp[i*6+k] * bk[k*6+j];
        }
    }
    Msp[t] = a; Mbp[t] = b;
}

// ------- zero-fill pad planes 6,7 of sOut/bOut (done once) ------------------
__global__ void k_zeropad(float* __restrict__ sOut, float* __restrict__ bOut) {
    int i = blockIdx.x * 256 + threadIdx.x;   // covers 2*NPIX
    sOut[6*NPIX + i] = 0.f;
    bOut[6*NPIX + i] = 0.f;
}

// ------- q0 = transpose(unaries): HWC -> CHW planar -------------------------
__global__ void k_initq(const float* __restrict__ un, float* __restrict__ q) {
    int pix = blockIdx.x * 256 + threadIdx.x;
    #pragma unroll
    for (int c = 0; c < CC; ++c) q[c*NPIX + pix] = un[pix*CC + c];
}

// ------- softmax over channels ----------------------------------------------
__global__ void k_softmax(const float* __restrict__ q, float* __restrict__ p) {
    int pix = blockIdx.x * 256 + threadIdx.x;
    float v[CC];
    float mx = -3.402823e38f;
    #pragma unroll
    for (int c = 0; c < CC; ++c) { v[c] = q[c*NPIX + pix]; mx = fmaxf(mx, v[c]); }
    float s = 0.f;
    #pragma unroll
    for (int c = 0; c < CC; ++c) { v[c] = expf(v[c] - mx); s += v[c]; }
    float inv = 1.f / s;
    #pragma unroll
    for (int c = 0; c < CC; ++c) p[c*NPIX + pix] = v[c] * inv;
}

// ------- separable Gaussian, pass along x (LDS row) -------------------------
__global__ void k_convx(const float* __restrict__ p, float* __restrict__ tmp) {
    __shared__ float row[WW + 2*RSP];
    int cy = blockIdx.x;            // c*HH + y
    int c = cy / HH, y = cy % HH;
    int x = threadIdx.x;
    row[x + RSP] = p[c*NPIX + y*WW + x];
    if (x < RSP) { row[x] = 0.f; row[WW + RSP + x] = 0.f; }
    __syncthreads();
    float s = 0.f;
    #pragma unroll
    for (int d = 0; d < 2*RSP+1; ++d) s += GK[d] * row[x + d];
    tmp[c*NPIX + y*WW + x] = s;
}

// ------- pass along y + normalization by filtered-ones ----------------------
__global__ void k_convy(const float* __restrict__ tmp, float* __restrict__ sOut) {
    int cy = blockIdx.x;
    int c = cy / HH, y = cy % HH;
    int x = threadIdx.x;
    float s = 0.f, ny = 0.f, nx = 0.f;
    #pragma unroll
    for (int d = 0; d < 2*RSP+1; ++d) {
        int yy = y + d - RSP;
        int xx = x + d - RSP;
        if (yy >= 0 && yy < HH) { s += GK[d] * tmp[c*NPIX + yy*WW + x]; ny += GK[d]; }
        if (xx >= 0 && xx < WW) nx += GK[d];
    }
    sOut[c*NPIX + y*WW + x] = s / (ny * nx);
}

// ------- bilateral: LDS tile via async-to-LDS, weights on the fly -----------
__global__ void k_bilateral(const float* __restrict__ rgb, const float* __restrict__ p,
                            float* __restrict__ bOut) {
    __shared__ float sR[BT*BT], sG[BT*BT], sB[BT*BT];
    __shared__ float sP[CC][BT*BT];
    int bx = blockIdx.x & 15, by = blockIdx.x >> 4;
    int x0 = bx * TILE, y0 = by * TILE;
    int tid = threadIdx.x;

    // Pre-fill with out-of-image defaults: huge rgb -> bilateral weight == 0,
    // p == 0 (zero pad). In-bounds entries get overwritten below.
    for (int i = tid; i < BT*BT; i += 256) {
        sR[i] = 1e8f; sG[i] = 1e8f; sB[i] = 1e8f;
        #pragma unroll
        for (int c = 0; c < CC; ++c) sP[c][i] = 0.f;
    }
    __syncthreads();

    for (int i = tid; i < BT*BT; i += 256) {
        int tx = i % BT, ty = i / BT;
        int gx = x0 - RBW + tx, gy = y0 - RBW + ty;
        if (gx >= 0 && gx < WW && gy >= 0 && gy < HH) {
            int gp = gy*WW + gx;
#ifdef HAVE_ASYNC_LDS
            __builtin_amdgcn_global_load_async_to_lds_b32(
                (int*)(rgb + gp*3 + 0), (int*)&sR[i], 0, 0);
            __builtin_amdgcn_global_load_async_to_lds_b32(
                (int*)(rgb + gp*3 + 1), (int*)&sG[i], 0, 0);
            __builtin_amdgcn_global_load_async_to_lds_b32(
                (int*)(rgb + gp*3 + 2), (int*)&sB[i], 0, 0);
            #pragma unroll
            for (int c = 0; c < CC; ++c)
                __builtin_amdgcn_global_load_async_to_lds_b32(
                    (int*)(p + c*NPIX + gp), (int*)&sP[c][i], 0, 0);
#else
            sR[i] = rgb[gp*3 + 0];
            sG[i] = rgb[gp*3 + 1];
            sB[i] = rgb[gp*3 + 2];
            #pragma unroll
            for (int c = 0; c < CC; ++c) sP[c][i] = p[c*NPIX + gp];
#endif
        }
    }
#ifdef HAVE_ASYNC_LDS
    __builtin_amdgcn_s_wait_asynccnt(0);
#endif
    __syncthreads();

    int lx = tid & 15, ly = tid >> 4;
    int ci = (ly + RBW)*BT + (lx + RBW);
    float cr = sR[ci], cg = sG[ci], cb = sB[ci];
    float wsum = 0.f;
    float acc[CC] = {0.f, 0.f, 0.f, 0.f, 0.f, 0.f};
    for (int dy = 0; dy < 2*RBW+1; ++dy) {
        for (int dx = 0; dx < 2*RBW+1; ++dx) {
            int ni = (ly + dy)*BT + (lx + dx);
            float dr = cr - sR[ni], dg = cg - sG[ni], db = cb - sB[ni];
            float d2 = dr*dr + dg*dg + db*db;
            float oy = (float)(dy - RBW), ox = (float)(dx - RBW);
            // theta_alpha=160 -> 1/(2*160^2); theta_beta=3 -> 1/18
            float w = expf(-(oy*oy + ox*ox) * (1.0f/51200.0f) - d2 * (1.0f/18.0f));
            wsum += w;
            #pragma unroll
            for (int c = 0; c < CC; ++c) acc[c] += w * sP[c][ni];
        }
    }
    int pix = (y0 + ly)*WW + (x0 + lx);
    float inv = 1.f / wsum;
    #pragma unroll
    for (int c = 0; c < CC; ++c) bOut[c*NPIX + pix] = acc[c] * inv;
}

// ------- superpixel segment sums --------------------------------------------
__global__ void k_segclear(float* __restrict__ segB) {
    segB[blockIdx.x * 256 + threadIdx.x] = 0.f;
}

__global__ void k_segaccum(const float* __restrict__ p, const int* __restrict__ spMap,
                           float* __restrict__ segB) {
    int pix = blockIdx.x * 256 + threadIdx.x;
    int y = pix >> 8, x = pix & 255;
    int seg = spMap[x*WW + y];   // reference's transposed lookup quirk
    #pragma unroll
    for (int c = 0; c < CC; ++c)
        atomicAdd(&segB[seg*8 + c], logf(p[c*NPIX + pix] + EPSF));
}

// ------- final: WMMA channel-mix + superpixel term + unary ------------------
// pairwise = Ms @ sOut + Mb @ bOut via V_WMMA_F32_16X16X4_F32
// (M = channels padded to 16, N = 16 pixels, K split 4+4 with zero pad).
// All operands are unconditionally loadable thanks to zero-padded planes/mats.
__global__ void k_final(const float* __restrict__ sOut, const float* __restrict__ bOut,
                        const float* __restrict__ p,
                        const float* __restrict__ Msp, const float* __restrict__ Mbp,
                        const float* __restrict__ segB, const int* __restrict__ spMap,
                        const float* __restrict__ un, const float* __restrict__ lowW,
                        const float* __restrict__ highW, float* __restrict__ qOut) {
    int tid  = blockIdx.x * blockDim.x + threadIdx.x;
    int wave = tid >> 5;
    int lane = threadIdx.x & 31;
    int col  = lane & 15;
    int half = lane >> 4;
    int p0   = wave * 32;           // 32 pixels per wave (2 WMMA column groups)

    // A operands, 16x4 f32 layout: lanes 0-15 hold K=0,1; lanes 16-31 K=2,3.
    // Msp/Mbp are 16 rows x 8 cols, zero padded -> no conditionals.
    int m = col;
    v2f_t a0s, a1s, a0b, a1b;
    a0s.x = Msp[m*8 + 2*half];      a0s.y = Msp[m*8 + 2*half + 1];
    a1s.x = Msp[m*8 + 4 + 2*half];  a1s.y = Msp[m*8 + 5 + 2*half];
    a0b.x = Mbp[m*8 + 2*half];      a0b.y = Mbp[m*8 + 2*half + 1];
    a1b.x = Mbp[m*8 + 4 + 2*half];  a1b.y = Mbp[m*8 + 5 + 2*half];

    v8f_t acc[2];
    #pragma unroll
    for (int g = 0; g < 2; ++g) {
        int pg = p0 + g*16 + col;
        // B operands, 4x16 layout (column = pixel); planes 6,7 are zeros.
        v2f_t b0s, b1s, b0b, b1b;
        b0s.x = sOut[(2*half + 0)*NPIX + pg];
        b0s.y = sOut[(2*half + 1)*NPIX + pg];
        b1s.x = sOut[(2*half + 4)*NPIX + pg];
        b1s.y = sOut[(2*half + 5)*NPIX + pg];
        b0b.x = bOut[(2*half + 0)*NPIX + pg];
        b0b.y = bOut[(2*half + 1)*NPIX + pg];
        b1b.x = bOut[(2*half + 4)*NPIX + pg];
        b1b.y = bOut[(2*half + 5)*NPIX + pg];
        v8f_t c = {0.f, 0.f, 0.f, 0.f, 0.f, 0.f, 0.f, 0.f};
        c = __builtin_amdgcn_wmma_f32_16x16x4_f32(false, a0s, false, b0s, (short)0, c, false, false);
        c = __builtin_amdgcn_wmma_f32_16x16x4_f32(false, a1s, false, b1s, (short)0, c, false, false);
        c = __builtin_amdgcn_wmma_f32_16x16x4_f32(false, a0b, false, b0b, (short)0, c, false, false);
        c = __builtin_amdgcn_wmma_f32_16x16x4_f32(false, a1b, false, b1b, (short)0, c, false, false);
        acc[g] = c;
    }

    // D layout: lanes 0-15 hold rows 0-7 (our 6 channels) for their column.
    // Move group-1 results to lanes 16-31 so every lane finalizes one pixel.
    int pix = p0 + lane;
    float pw[CC];
    #pragma unroll
    for (int c = 0; c < CC; ++c) {
        float vB = __shfl(acc[1][c], col, 32);
        pw[c] = (half == 0) ? acc[0][c] : vB;
    }

    int y = pix >> 8, x = pix & 255;
    int seg = spMap[x*WW + y];
    float hw = highW[0];
    #pragma unroll
    for (int c = 0; c < CC; ++c) {
        float pv = p[c*NPIX + pix];
        float first = expf(segB[seg*8 + c] - logf(pv + EPSF));
        float spu = lowW[c] * first + hw * (1.f - first);
        qOut[c*NPIX + pix] = un[pix*CC + c] - pw[c] - spu;
    }
}

// ------- output: CHW planar -> (1,H,W,C) ------------------------------------
__global__ void k_writeout(const float* __restrict__ q, float* __restrict__ out) {
    int pix = blockIdx.x * 256 + threadIdx.x;
    #pragma unroll
    for (int c = 0; c < CC; ++c) out[pix*CC + c] = q[c*NPIX + pix];
}

extern "C" void kernel_launch(void* const* d_in, const int* in_sizes, int n_in,
                              void* d_out, int out_size, void* d_ws, size_t ws_size,
                              hipStream_t stream) {
    (void)in_sizes; (void)n_in; (void)out_size; (void)ws_size;
    const float* un   = (const float*)d_in[0];   // (1,H,W,C)
    const float* rgb  = (const float*)d_in[1];   // (1,H,W,3)
    const int*   sp   = (const int*)  d_in[2];   // (1,H,W)
    const float* sk   = (const float*)d_in[3];   // 6x6
    const float* bk   = (const float*)d_in[4];   // 6x6
    const float* comp = (const float*)d_in[5];   // 6x6
    const float* lw   = (const float*)d_in[6];   // 6
    const float* hw   = (const float*)d_in[7];   // 1
    float* out = (float*)d_out;

    float* ws   = (float*)d_ws;
    float* p    = ws;                 // 6*N
    float* tmp  = ws + 6*NPIX;        // 6*N
    float* sOut = ws + 12*NPIX;       // 8*N (planes 6,7 zero)
    float* bOut = ws + 20*NPIX;       // 8*N (planes 6,7 zero)
    float* q    = ws + 28*NPIX;       // 6*N
    float* segB = ws + 34*NPIX;       // 512*8 padded
    float* Msp  = segB + 4096;        // 16x8 padded
    float* Mbp  = Msp + 128;          // 16x8 padded

    k_prep<<<1, 128, 0, stream>>>(sk, bk, comp, Msp, Mbp);
    k_zeropad<<<2*NPIX/256, 256, 0, stream>>>(sOut, bOut);
    k_initq<<<NPIX/256, 256, 0, stream>>>(un, q);
    for (int it = 0; it < 5; ++it) {
        k_softmax  <<<NPIX/256, 256, 0, stream>>>(q, p);
        k_convx    <<<CC*HH,    256, 0, stream>>>(p, tmp);
        k_convy    <<<CC*HH,    256, 0, stream>>>(tmp, sOut);
        k_bilateral<<<256,      256, 0, stream>>>(rgb, p, bOut);
        k_segclear <<<16,       256, 0, stream>>>(segB);
        k_segaccum <<<NPIX/256, 256, 0, stream>>>(p, sp, segB);
        k_final    <<<NPIX/256, 256, 0, stream>>>(sOut, bOut, p, Msp, Mbp, segB,
                                                  sp, un, lw, hw, q);
    }
    k_writeout<<<NPIX/256, 256, 0, stream>>>(q, out);
}